// CF2DLayer_15796889715293
// MI455X (gfx1250) — compile-verified
//
#include <hip/hip_runtime.h>

typedef __attribute__((ext_vector_type(2))) float v2f;
typedef __attribute__((ext_vector_type(8))) float v8f;

__device__ __forceinline__ v8f wmma_f32(v2f a, v2f b, v8f c) {
  // D = A(16x4) * B(4x16) + C, fp32 WMMA
  return __builtin_amdgcn_wmma_f32_16x16x4_f32(false, a, false, b, (short)0, c, false, false);
}

// A-fragment: row-major A[m][k], row stride ld (floats).
// lane<16: m=m0+lane, v0=A[m][k0],   v1=A[m][k0+1]
// lane>=16:            v0=A[m][k0+2], v1=A[m][k0+3]
template <typename PT>
__device__ __forceinline__ v2f frag_a(const PT* A, int ld, int m0, int k0, int lane) {
  int m = m0 + (lane & 15);
  int k = k0 + ((lane >> 4) << 1);
  v2f r; r.x = A[m * ld + k]; r.y = A[m * ld + k + 1]; return r;
}

// B-fragment: row-major B[k][n], row stride ld.
template <typename PT>
__device__ __forceinline__ v2f frag_b(const PT* Bm, int ld, int k0, int n0, int lane) {
  int n = n0 + (lane & 15);
  int k = k0 + ((lane >> 4) << 1);
  v2f r; r.x = Bm[k * ld + n]; r.y = Bm[(k + 1) * ld + n]; return r;
}

// C/D layout: VGPR v, lanes 0-15: M=m0+v, lanes 16-31: M=m0+8+v; N = n0 + (lane&15)
__device__ __forceinline__ void store_c(float* out, int ld, int m0, int n0, int lane, v8f c) {
  int n = n0 + (lane & 15);
  int mb = m0 + ((lane >> 4) << 3);
#pragma unroll
  for (int v = 0; v < 8; ++v) out[(size_t)(mb + v) * ld + n] = c[v];
}

// ---------- init: build transform matrices (double precision trig) ----------
__global__ void init_consts(float* ws) {
  int idx = blockIdx.x * 256 + threadIdx.x;
  if (idx >= 6 * 8192) return;
  int region = idx >> 13;
  int r = idx & 8191;
  float* Dm = ws;             // [b][n] 32x256  (truncated DCT rows)
  float* Dt = ws + 8192;      // [n][b] 256x32  (DCT^T)
  float* Fr = ws + 16384;     // [a][x] 32x256  cos(-2pi a x/N)/16
  float* Fi = ws + 24576;     // [a][x]         -sin(2pi a x/N)/16
  float* Cr = ws + 32768;     // [X][a] 256x32  g_a cos/16
  float* Ci = ws + 40960;     // [X][a]        -g_a sin/16
  const double PI = 3.14159265358979323846;
  if (region == 0) {
    int b = r >> 8, n = r & 255;
    double s = (b == 0) ? sqrt(1.0 / 256.0) : sqrt(2.0 / 256.0);
    Dm[r] = (float)(cos(PI * (n + 0.5) * b / 256.0) * s);
  } else if (region == 1) {
    int n = r >> 5, b = r & 31;
    double s = (b == 0) ? sqrt(1.0 / 256.0) : sqrt(2.0 / 256.0);
    Dt[r] = (float)(cos(PI * (n + 0.5) * b / 256.0) * s);
  } else if (region == 2) {
    int a = r >> 8, x = r & 255;
    Fr[r] = (float)(cos(2.0 * PI * a * x / 256.0) / 16.0);
  } else if (region == 3) {
    int a = r >> 8, x = r & 255;
    Fi[r] = (float)(-sin(2.0 * PI * a * x / 256.0) / 16.0);
  } else if (region == 4) {
    int X = r >> 5, a = r & 31;
    double g = (a == 0) ? 1.0 : 2.0;
    Cr[r] = (float)(g * cos(2.0 * PI * a * X / 256.0) / 16.0);
  } else {
    int X = r >> 5, a = r & 31;
    double g = (a == 0) ? 1.0 : 2.0;
    Ci[r] = (float)(-g * sin(2.0 * PI * a * X / 256.0) / 16.0);
  }
}

// ---------- transpose R: view (i,j,a,b) as 1024x1024 [ij][ab] -> [ab][ij] ----------
__global__ __launch_bounds__(256) void transpose1024(const float* __restrict__ Rr,
                                                     const float* __restrict__ Ri,
                                                     float* __restrict__ RTr,
                                                     float* __restrict__ RTi) {
  __shared__ float tile[32][33];
  const float* src = blockIdx.z ? Ri : Rr;
  float* dst = blockIdx.z ? RTi : RTr;
  int bx = blockIdx.x * 32, by = blockIdx.y * 32;
  int tx = threadIdx.x, ty = threadIdx.y;  // 32 x 8
  for (int r = ty; r < 32; r += 8)
    tile[r][tx] = src[(size_t)(by + r) * 1024 + bx + tx];
  __syncthreads();
  for (int r = ty; r < 32; r += 8)
    dst[(size_t)(bx + r) * 1024 + by + tx] = tile[tx][r];
}

// ---------- stage A: t1[e,X,b,c] = sum_n Dm[b,n] * x[e,X,n,c] ----------
__global__ __launch_bounds__(128) void stageA(const float* __restrict__ x,
                                              const float* __restrict__ Dm,
                                              float* __restrict__ t1) {
  __shared__ float sD[32 * 65];  // [m][kk]
  __shared__ float sX[64 * 33];  // [kk][c]
  int eX = blockIdx.x;  // e*256+X
  int tid = threadIdx.x, lane = tid & 31, wave = tid >> 5;
  int tb = (wave >> 1) * 16, tc = (wave & 1) * 16;
  const float* xs = x + (size_t)eX * (256 * 32);
  v8f acc = {};
  for (int k0 = 0; k0 < 256; k0 += 64) {
    for (int i = tid; i < 32 * 64; i += 128) {
      int m = i >> 6, kk = i & 63;
      sD[m * 65 + kk] = Dm[m * 256 + k0 + kk];
    }
    for (int i = tid; i < 64 * 32; i += 128) {
      int kk = i >> 5, c = i & 31;
      sX[kk * 33 + c] = xs[(size_t)(k0 + kk) * 32 + c];
    }
    __syncthreads();
#pragma unroll
    for (int kk = 0; kk < 64; kk += 4)
      acc = wmma_f32(frag_a(sD, 65, tb, kk, lane), frag_b(sX, 33, kk, tc, lane), acc);
    __syncthreads();
  }
  store_c(t1 + (size_t)eX * 1024, 32, tb, tc, lane, acc);
}

// ---------- stage B: vt[e,a,b,c] = sum_X F[a,X] * t1[e,X,b,c] (real & imag) ----------
__global__ __launch_bounds__(256) void stageB(const float* __restrict__ t1,
                                              const float* __restrict__ Fr,
                                              const float* __restrict__ Fi,
                                              float* __restrict__ vtr,
                                              float* __restrict__ vti) {
  __shared__ float sFr[32 * 65];
  __shared__ float sFi[32 * 65];
  __shared__ float sT[64 * 33];
  int e = blockIdx.x >> 5, b = blockIdx.x & 31;
  int tid = threadIdx.x, lane = tid & 31, wave = tid >> 5;
  int imag = wave >> 2, th = wave & 3;
  int ta = (th >> 1) * 16, tc = (th & 1) * 16;
  v8f acc = {};
  for (int k0 = 0; k0 < 256; k0 += 64) {
    for (int i = tid; i < 32 * 64; i += 256) {
      int m = i >> 6, kk = i & 63;
      sFr[m * 65 + kk] = Fr[m * 256 + k0 + kk];
      sFi[m * 65 + kk] = Fi[m * 256 + k0 + kk];
    }
    for (int i = tid; i < 64 * 32; i += 256) {
      int kk = i >> 5, c = i & 31;
      sT[kk * 33 + c] = t1[(((size_t)e * 256 + k0 + kk) * 32 + b) * 32 + c];
    }
    __syncthreads();
    const float* sF = imag ? sFi : sFr;
#pragma unroll
    for (int kk = 0; kk < 64; kk += 4)
      acc = wmma_f32(frag_a(sF, 65, ta, kk, lane), frag_b(sT, 33, kk, tc, lane), acc);
    __syncthreads();
  }
  float* out = (imag ? vti : vtr) + ((size_t)e * 1024 + b) * 32;
  store_c(out, 1024, ta, tc, lane, acc);
}

// ---------- stage C: per mode (a,b): E[e,i] = sum_j V[e,j] * R[i,j] (complex) ----------
__global__ __launch_bounds__(128) void stageC(const float* __restrict__ vtr,
                                              const float* __restrict__ vti,
                                              const float* __restrict__ RTr,
                                              const float* __restrict__ RTi,
                                              float* __restrict__ er,
                                              float* __restrict__ ei) {
  __shared__ float sVr[16 * 33], sVi[16 * 33];
  __shared__ float sRr[32 * 33], sRi[32 * 33], sRin[32 * 33];
  int ab = blockIdx.x;
  int tid = threadIdx.x, lane = tid & 31, wave = tid >> 5;
  for (int i = tid; i < 16 * 32; i += 128) {
    int e = i >> 5, j = i & 31;
    size_t off = ((size_t)e * 1024 + ab) * 32 + j;
    sVr[e * 33 + j] = vtr[off];
    sVi[e * 33 + j] = vti[off];
  }
  for (int i = tid; i < 32 * 32; i += 128) {
    int ii = i >> 5, j = i & 31;
    float rr = RTr[(size_t)ab * 1024 + i];
    float rv = RTi[(size_t)ab * 1024 + i];
    sRr[ii * 33 + j] = rr;
    sRi[ii * 33 + j] = rv;
    sRin[ii * 33 + j] = -rv;
  }
  __syncthreads();
  int imag = wave >> 1, ti = (wave & 1) * 16;
  v8f acc = {};
#pragma unroll
  for (int k = 0; k < 32; k += 4) {
    if (imag) {
      // Ei = Vr*Ri^T + Vi*Rr^T   (B[k=j][n=i] = R[i][j] -> frag_a on row-major R)
      acc = wmma_f32(frag_a(sVr, 33, 0, k, lane), frag_a(sRi, 33, ti, k, lane), acc);
      acc = wmma_f32(frag_a(sVi, 33, 0, k, lane), frag_a(sRr, 33, ti, k, lane), acc);
    } else {
      // Er = Vr*Rr^T - Vi*Ri^T
      acc = wmma_f32(frag_a(sVr, 33, 0, k, lane), frag_a(sRr, 33, ti, k, lane), acc);
      acc = wmma_f32(frag_a(sVi, 33, 0, k, lane), frag_a(sRin, 33, ti, k, lane), acc);
    }
  }
  float* out = (imag ? ei : er) + (size_t)ab * 32;  // layout (e,a,b,i), ld_e = 32768
  store_c(out, 32768, 0, ti, lane, acc);
}

// ---------- stage D: t3[e,i,X,b] = sum_a Cr[X,a]*Er[e,a,b,i] + Ci[X,a]*Ei[e,a,b,i] ----------
__global__ __launch_bounds__(256) void stageD(const float* __restrict__ er,
                                              const float* __restrict__ ei,
                                              const float* __restrict__ Cr,
                                              const float* __restrict__ Ci,
                                              float* __restrict__ t3) {
  __shared__ float sEr[32 * 33], sEi[32 * 33];  // [a][b]
  int e = blockIdx.x >> 5, ich = blockIdx.x & 31;
  int tid = threadIdx.x, lane = tid & 31, wave = tid >> 5;
  for (int i = tid; i < 32 * 32; i += 256) {
    int aa = i >> 5, bb = i & 31;
    size_t off = (size_t)e * 32768 + (size_t)aa * 1024 + (size_t)bb * 32 + ich;
    sEr[aa * 33 + bb] = er[off];
    sEi[aa * 33 + bb] = ei[off];
  }
  __syncthreads();
  float* out = t3 + ((size_t)e * 32 + ich) * 8192;  // [X][b], ld 32
#pragma unroll
  for (int t = 0; t < 4; ++t) {
    int tile = wave * 4 + t;
    int tx = (tile >> 1) * 16, tb = (tile & 1) * 16;
    v8f acc = {};
#pragma unroll
    for (int k = 0; k < 32; k += 4) {
      acc = wmma_f32(frag_a(Cr, 32, tx, k, lane), frag_b(sEr, 33, k, tb, lane), acc);
      acc = wmma_f32(frag_a(Ci, 32, tx, k, lane), frag_b(sEi, 33, k, tb, lane), acc);
    }
    store_c(out, 32, tx, tb, lane, acc);
  }
}

// ---------- stage E: out[e,X,N,i] = relu( sum_b Dt[N,b]*t3[e,i,X,b] + sum_c x[e,X,N,c]*W[i,c] + bias[i] ) ----------
__global__ __launch_bounds__(256) void stageE(const float* __restrict__ x,
                                              const float* __restrict__ t3,
                                              const float* __restrict__ Dt,
                                              const float* __restrict__ W,
                                              const float* __restrict__ bias,
                                              float* __restrict__ out) {
  __shared__ float sXm[256 * 33];  // [N][c]
  __shared__ float sS[32 * 33];    // [b][i]
  int eX = blockIdx.x;
  int e = eX >> 8, X = eX & 255;
  int tid = threadIdx.x, lane = tid & 31, wave = tid >> 5;
  const float* xs = x + (size_t)eX * (256 * 32);
  for (int i = tid; i < 256 * 32; i += 256) {
    int n = i >> 5, c = i & 31;
    sXm[n * 33 + c] = xs[i];
  }
  for (int i = tid; i < 32 * 32; i += 256) {
    int ii = i >> 5, bb = i & 31;  // bb fastest -> coalesced 128B rows of t3
    sS[bb * 33 + ii] = t3[((size_t)e * 32 + ii) * 8192 + (size_t)X * 32 + bb];
  }
  __syncthreads();
  float* os = out + (size_t)eX * (256 * 32);
#pragma unroll
  for (int t = 0; t < 4; ++t) {
    int tile = wave * 4 + t;
    int tn = (tile >> 1) * 16, ti = (tile & 1) * 16;
    float bv = bias[ti + (lane & 15)];
    v8f acc;
#pragma unroll
    for (int v = 0; v < 8; ++v) acc[v] = bv;
#pragma unroll
    for (int k = 0; k < 32; k += 4) {
      acc = wmma_f32(frag_a(Dt, 32, tn, k, lane), frag_b(sS, 33, k, ti, lane), acc);
      // B[k=c][n=i] = W[i][c] -> frag_a on row-major W
      acc = wmma_f32(frag_a(sXm, 33, tn, k, lane), frag_a(W, 32, ti, k, lane), acc);
    }
    int nn = ti + (lane & 15);
    int mb = tn + ((lane >> 4) << 3);
#pragma unroll
    for (int v = 0; v < 8; ++v) {
      float val = acc[v];
      os[(size_t)(mb + v) * 32 + nn] = val > 0.f ? val : 0.f;
    }
  }
}

extern "C" void kernel_launch(void* const* d_in, const int* in_sizes, int n_in,
                              void* d_out, int out_size, void* d_ws, size_t ws_size,
                              hipStream_t stream) {
  (void)in_sizes; (void)n_in; (void)out_size; (void)ws_size;
  const float* x  = (const float*)d_in[0];
  const float* Rr = (const float*)d_in[1];
  const float* Ri = (const float*)d_in[2];
  const float* W  = (const float*)d_in[3];
  const float* b  = (const float*)d_in[4];

  float* ws  = (float*)d_ws;
  float* Dm  = ws;                  // 8192
  float* Fr  = ws + 16384;          // 8192
  float* Fi  = ws + 24576;          // 8192
  float* Cr  = ws + 32768;          // 8192
  float* Ci  = ws + 40960;          // 8192
  float* Dt  = ws + 8192;           // 8192
  float* vtr = ws + 49152;          // 524288
  float* vti = vtr + 524288;        // 524288
  float* er  = vti + 524288;        // 524288
  float* ei  = er + 524288;         // 524288
  float* RTr = ei + 524288;         // 1048576
  float* RTi = RTr + 1048576;       // 1048576
  float* t1  = RTi + 1048576;       // 4194304 (aliased as t3 after stage B)
  float* t3  = t1;

  init_consts<<<192, 256, 0, stream>>>(ws);
  dim3 tg(32, 32, 2), tb(32, 8, 1);
  transpose1024<<<tg, tb, 0, stream>>>(Rr, Ri, RTr, RTi);
  stageA<<<16 * 256, 128, 0, stream>>>(x, Dm, t1);
  stageB<<<16 * 32, 256, 0, stream>>>(t1, Fr, Fi, vtr, vti);
  stageC<<<1024, 128, 0, stream>>>(vtr, vti, RTr, RTi, er, ei);
  stageD<<<16 * 32, 256, 0, stream>>>(er, ei, Cr, Ci, t3);
  stageE<<<16 * 256, 256, 0, stream>>>(x, t3, Dt, W, b, (float*)d_out);
}